// TokenizerTranslator_18124761989510
// MI455X (gfx1250) — compile-verified
//
#include <hip/hip_runtime.h>

// Problem constants (from the reference)
#define S_    8
#define P_    2048
#define VSRC  8192
#define VTGT  8000
#define P_OUT (P_ - 2)   // 2046 rows after stripping prefix/suffix

// Native clang vector types (HIP_vector_type structs are rejected by the
// __builtin_nontemporal_* builtins; ext_vector_type is accepted).
typedef float __attribute__((ext_vector_type(4))) f32x4;
typedef int   __attribute__((ext_vector_type(4))) i32x4;

// ---------------------------------------------------------------------------
// Kernel 1: initialize the column->source map to -1 (unmapped columns -> 0.0f)
// ---------------------------------------------------------------------------
__global__ void tt_init_map(int* __restrict__ map) {
    int i = blockIdx.x * blockDim.x + threadIdx.x;
    if (i < VTGT) map[i] = -1;
}

// ---------------------------------------------------------------------------
// Kernel 2: build map from the 0/1 convert matrix (row-major VSRC x VTGT).
// Each column has at most one nonzero -> plain stores, no atomics.
// Read as f32x4 (VTGT % 4 == 0, so a vector never straddles rows).
// Single-use stream -> non-temporal loads to keep L2 clean.
// ---------------------------------------------------------------------------
__global__ void tt_build_map(const f32x4* __restrict__ conv4,
                             int* __restrict__ map) {
    long long i = (long long)blockIdx.x * blockDim.x + threadIdx.x; // vec4 idx
    const long long n4 = (long long)VSRC * VTGT / 4;
    if (i >= n4) return;
    f32x4 c = __builtin_nontemporal_load(&conv4[i]);
    long long e = i * 4;                 // flat element index
    int v = (int)(e / VTGT);
    int w = (int)(e - (long long)v * VTGT);
    if (c.x != 0.0f) map[w + 0] = v;
    if (c.y != 0.0f) map[w + 1] = v;
    if (c.z != 0.0f) map[w + 2] = v;
    if (c.w != 0.0f) map[w + 3] = v;
}

// ---------------------------------------------------------------------------
// Kernel 3: per output row, stage the 32KB source row into LDS with CDNA5
// async global->LDS B128 copies (ASYNCcnt path), then gather through LDS
// (i32x4 map reads, 4x ds_load, b128 non-temporal output stores).
// ---------------------------------------------------------------------------
__global__ __launch_bounds__(1024, 1) void tt_gather_rows(
    const float* __restrict__ x,
    const int*   __restrict__ map,
    float*       __restrict__ out) {

    __shared__ __align__(16) float row[VSRC];   // 32 KB

    const int bid = blockIdx.x;                 // = s * P_OUT + pp
    const int s   = bid / P_OUT;
    const int pp  = bid - s * P_OUT;
    const float* src = x + ((long long)s * P_ + (pp + 1)) * VSRC;

    const int t = threadIdx.x;

    // Generic->LDS address: low 32 bits of the generic pointer are the
    // wave-relative LDS byte offset (ISA 10.2).
    unsigned lds_base = (unsigned)(unsigned long long)(const void*)row;

    // 8192 floats = 1024 threads * 2 iters * 4 floats (b128 per lane).
    // Each async b128 moves 16B/lane -> 512B per wave32 instruction,
    // fully coalesced, tracked on ASYNCcnt (no VGPR landing zone).
#pragma unroll
    for (int k = 0; k < 2; ++k) {
        int elem = (t + k * 1024) * 4;                         // float index
        unsigned long long gaddr = (unsigned long long)(src + elem);
        unsigned laddr = lds_base + (unsigned)elem * 4u;       // byte offset
        asm volatile("global_load_async_to_lds_b128 %0, %1, off"
                     :: "v"(laddr), "v"(gaddr)
                     : "memory");
    }

#if __has_builtin(__builtin_amdgcn_s_wait_asynccnt)
    __builtin_amdgcn_s_wait_asynccnt(0);
#else
    asm volatile("s_wait_asynccnt 0" ::: "memory");
#endif
    __syncthreads();   // make all waves' async-filled LDS visible to everyone

    // Gather 4 columns per thread: i32x4 map load (L2-resident, regular
    // temporal), 4 LDS reads, one b128 non-temporal store (pure stream).
    const i32x4* map4 = (const i32x4*)map;
    f32x4*       dst4 = (f32x4*)(out + (long long)bid * VTGT);
#pragma unroll 2
    for (int w4 = t; w4 < VTGT / 4; w4 += 1024) {
        i32x4 m = map4[w4];
        f32x4 r;
        r.x = (m.x >= 0) ? row[m.x] : 0.0f;
        r.y = (m.y >= 0) ? row[m.y] : 0.0f;
        r.z = (m.z >= 0) ? row[m.z] : 0.0f;
        r.w = (m.w >= 0) ? row[m.w] : 0.0f;
        __builtin_nontemporal_store(r, &dst4[w4]);
    }
}

// ---------------------------------------------------------------------------
// Host-side launcher
// ---------------------------------------------------------------------------
extern "C" void kernel_launch(void* const* d_in, const int* in_sizes, int n_in,
                              void* d_out, int out_size, void* d_ws, size_t ws_size,
                              hipStream_t stream) {
    (void)in_sizes; (void)n_in; (void)out_size; (void)ws_size;

    const float* x    = (const float*)d_in[0];   // (S, P, VSRC) f32
    const float* conv = (const float*)d_in[1];   // (VSRC, VTGT) f32 0/1
    float*       out  = (float*)d_out;           // (S, P-2, VTGT) f32
    int*         map  = (int*)d_ws;              // VTGT ints of scratch

    // 1) map[w] = -1
    tt_init_map<<<(VTGT + 255) / 256, 256, 0, stream>>>(map);

    // 2) scan convert once (vec4-coalesced, NT) to build column->row map
    const long long n4 = (long long)VSRC * VTGT / 4;          // 16,384,000
    tt_build_map<<<(unsigned)((n4 + 255) / 256), 256, 0, stream>>>(
        (const f32x4*)conv, map);

    // 3) one workgroup per output row: async-stage row to LDS, gather, store
    tt_gather_rows<<<S_ * P_OUT, 1024, 0, stream>>>(x, map, out);
}